// _JoinerBiasing_5497558139009
// MI455X (gfx1250) — compile-verified
//
#include <hip/hip_runtime.h>
#include <hip/hip_bf16.h>

typedef __attribute__((ext_vector_type(2))) float v2f;
typedef __attribute__((ext_vector_type(4))) float v4f;
typedef __attribute__((ext_vector_type(8))) float v8f;

// Problem dimensions (fixed by the reference's setup_inputs()).
static constexpr int Bb = 8;
static constexpr int Tt = 256;
static constexpr int Uu = 64;
static constexpr int Dd = 512;   // joint dim
static constexpr int Aa = 256;   // biasing dim
static constexpr int Vv = 1024;  // vocab
static constexpr long long Mrows = (long long)Bb * Tt * Uu;  // 131072

static constexpr int KC   = 64;  // K chunk staged in LDS
static constexpr int BPAD = 68;  // LDS column stride (floats); bank = 4n+k, conflict-free

// D = A(16x4,f32) x B(4x16,f32) + C(16x16,f32), wave32.
__device__ __forceinline__ v8f wmma_f32(v2f a, v2f b, v8f c) {
    return __builtin_amdgcn_wmma_f32_16x16x4_f32(
        false, a, false, b, (short)0, c, false, false);
}

// ---------------------------------------------------------------------------
// Block = 128 threads (4 waves). Block tile: 64(M) x 64(N).
// Wave w owns rows [row0 + w*16, +16); all waves share the KxN weight panel,
// staged through double-buffered LDS with a one-chunk pipeline.
// Weight matrix is (N_total, K) row-major => contiguous along K.
// ---------------------------------------------------------------------------
#define GEMM_BODY(KDIM, APTR, ALD, WPTR, WLD)                                   \
    __shared__ float Bsh[2][64 * BPAD];                                         \
    const int tid   = threadIdx.x;                                              \
    const int lane  = tid & 31;                                                 \
    const int wave  = tid >> 5;                                                 \
    const int n0    = blockIdx.x * 64;                                          \
    const long long rowblk = (long long)blockIdx.y * 64;                        \
    const long long row0   = rowblk + wave * 16;                                \
    const int nl    = lane & 15;                                                \
    const int khalf = (lane >> 4) * 2;                                          \
    const int mhalf = (lane >> 4) * 8;                                          \
    /* cooperative B-panel mapping: 4096 floats/chunk = 8 x float4 per thread */\
    int bcol[8], bkk[8];                                                        \
    _Pragma("unroll") for (int i = 0; i < 8; ++i) {                             \
        int e  = tid + i * 128;                                                 \
        bcol[i] = e >> 4;                                                       \
        bkk[i]  = (e & 15) * 4;                                                 \
    }                                                                           \
    v4f breg[8];                                                                \
    /* chunk 0 -> LDS buffer 0 */                                               \
    _Pragma("unroll") for (int i = 0; i < 8; ++i)                               \
        breg[i] = *(const v4f*)(WPTR + (long long)(n0 + bcol[i]) * WLD + bkk[i]);\
    _Pragma("unroll") for (int i = 0; i < 8; ++i)                               \
        *(v4f*)(&Bsh[0][bcol[i] * BPAD + bkk[i]]) = breg[i];                    \
    __syncthreads();                                                            \
    v8f acc0 = {}, acc1 = {}, acc2 = {}, acc3 = {};                             \
    const float* ap = APTR + (row0 + nl) * ALD + khalf;                         \
    v2f a_cur = *(const v2f*)(ap);                                              \
    const int nchunks = KDIM / KC;                                              \
    for (int c = 0; c < nchunks; ++c) {                                         \
        const int kbase = c * KC;                                               \
        if (c + 1 < nchunks) { /* prefetch next weight chunk into registers */  \
            _Pragma("unroll") for (int i = 0; i < 8; ++i)                       \
                breg[i] = *(const v4f*)(WPTR + (long long)(n0 + bcol[i]) * WLD  \
                                        + (kbase + KC) + bkk[i]);               \
        }                                                                       \
        const float* bs = &Bsh[c & 1][0];                                       \
        _Pragma("unroll 4") for (int k = 0; k < KC; k += 4) {                   \
            int kn = kbase + k + 4;                                             \
            if (kn > KDIM - 4) kn = KDIM - 4; /* harmless clamp at tail */      \
            v2f a_nxt = *(const v2f*)(ap + kn);                                 \
            v2f b0 = *(const v2f*)(bs + (0 * 16 + nl) * BPAD + khalf + k);      \
            v2f b1 = *(const v2f*)(bs + (1 * 16 + nl) * BPAD + khalf + k);      \
            v2f b2 = *(const v2f*)(bs + (2 * 16 + nl) * BPAD + khalf + k);      \
            v2f b3 = *(const v2f*)(bs + (3 * 16 + nl) * BPAD + khalf + k);      \
            acc0 = wmma_f32(a_cur, b0, acc0);                                   \
            acc1 = wmma_f32(a_cur, b1, acc1);                                   \
            acc2 = wmma_f32(a_cur, b2, acc2);                                   \
            acc3 = wmma_f32(a_cur, b3, acc3);                                   \
            a_cur = a_nxt;                                                      \
        }                                                                       \
        if (c + 1 < nchunks) { /* stage prefetched chunk; 1 barrier / chunk */  \
            _Pragma("unroll") for (int i = 0; i < 8; ++i)                       \
                *(v4f*)(&Bsh[(c + 1) & 1][bcol[i] * BPAD + bkk[i]]) = breg[i];  \
            __syncthreads();                                                    \
        }                                                                       \
    }

// ---------------------------------------------------------------------------
// Kernel 1: act = relu( hptr(M x A) @ W_bias^T + src + tgt + b_bias )
// rowblk is a multiple of 64 == U, so the block covers exactly one (b, t)
// with u = wave*16 + m; the src term is per-column.
// ---------------------------------------------------------------------------
__global__ __launch_bounds__(128)
void joiner_bias_relu_kernel(const float* __restrict__ hptr,
                             const float* __restrict__ Wb,   // (D, A)
                             const float* __restrict__ bb,   // (D)
                             const float* __restrict__ src,  // (B, T, D)
                             const float* __restrict__ tgt,  // (B, U, D)
                             float* __restrict__ act)        // (M, D)
{
    GEMM_BODY(Aa, hptr, Aa, Wb, Aa)

    const long long TU  = (long long)Tt * Uu;
    const long long bi  = rowblk / TU;
    const long long rem = rowblk - bi * TU;
    const int t = (int)(rem / Uu);      // rem % U == 0 (64 | rowblk)

    const float* srow  = src + (bi * Tt + t) * Dd;
    const float* tbase = tgt + bi * Uu * Dd;

    v8f accs[4] = {acc0, acc1, acc2, acc3};
#pragma unroll
    for (int s = 0; s < 4; ++s) {
        const int d = n0 + s * 16 + nl;
        const float sv = srow[d] + bb[d];
#pragma unroll
        for (int r = 0; r < 8; ++r) {
            const int m = r + mhalf;
            const int u = wave * 16 + m;
            float v = accs[s][r] + sv + tbase[(long long)u * Dd + d];
            v = fmaxf(v, 0.0f);
            act[(row0 + m) * Dd + d] = v;
        }
    }
}

// ---------------------------------------------------------------------------
// Kernel 2: out = act(M x D) @ W_out^T + b_out
// blockIdx.x = vocab block (fastest) so all 16 vocab blocks of an M-supertile
// dispatch together and the activation rows stay hot in L2.
// ---------------------------------------------------------------------------
__global__ __launch_bounds__(128)
void joiner_out_kernel(const float* __restrict__ act,  // (M, D)
                       const float* __restrict__ Wo,   // (V, D)
                       const float* __restrict__ bo,   // (V)
                       float* __restrict__ out)        // (M, V)
{
    GEMM_BODY(Dd, act, Dd, Wo, Dd)

    v8f accs[4] = {acc0, acc1, acc2, acc3};
#pragma unroll
    for (int s = 0; s < 4; ++s) {
        const int v = n0 + s * 16 + nl;
        const float bv = bo[v];
#pragma unroll
        for (int r = 0; r < 8; ++r) {
            const int m = r + mhalf;
            out[(row0 + m) * Vv + v] = accs[s][r] + bv;
        }
    }
}

// Lengths pass-through (int32 -> float in the concatenated float output).
__global__ void copy_lengths_kernel(const int* __restrict__ sl,
                                    const int* __restrict__ tl,
                                    float* __restrict__ out_sl,
                                    float* __restrict__ out_tl)
{
    int i = threadIdx.x;
    if (i < Bb) {
        out_sl[i] = (float)sl[i];
        out_tl[i] = (float)tl[i];
    }
}

extern "C" void kernel_launch(void* const* d_in, const int* in_sizes, int n_in,
                              void* d_out, int out_size, void* d_ws, size_t ws_size,
                              hipStream_t stream) {
    (void)in_sizes; (void)n_in; (void)out_size; (void)d_ws; (void)ws_size;

    const float* src  = (const float*)d_in[0];  // (B,T,D)
    const int*   slen = (const int*)  d_in[1];  // (B,)
    const float* tgt  = (const float*)d_in[2];  // (B,U,D)
    const int*   tlen = (const int*)  d_in[3];  // (B,)
    const float* hptr = (const float*)d_in[4];  // (B,T,U,A)
    const float* Wb   = (const float*)d_in[5];  // (D,A)
    const float* bb   = (const float*)d_in[6];  // (D,)
    const float* Wo   = (const float*)d_in[7];  // (V,D)
    const float* bo   = (const float*)d_in[8];  // (V,)

    float* out    = (float*)d_out;               // (B,T,U,V)
    float* out_sl = out + Mrows * Vv;            // (B,)
    float* out_tl = out_sl + Bb;                 // (B,)
    float* act    = out_tl + Bb;                 // (B,T,U,D) output #4

    const int msuper = (int)(Mrows / 64);        // 2048 block rows

    // Stage 1: biasing GEMM + broadcast-add + ReLU -> activation_out region.
    joiner_bias_relu_kernel<<<dim3(Dd / 64, msuper), 128, 0, stream>>>(
        hptr, Wb, bb, src, tgt, act);

    // Lengths (independent, tiny).
    copy_lengths_kernel<<<1, 32, 0, stream>>>(slen, tlen, out_sl, out_tl);

    // Stage 2: output GEMM reading activation_out (stream-ordered after stage 1).
    joiner_out_kernel<<<dim3(Vv / 64, msuper), 128, 0, stream>>>(
        act, Wo, bo, out);
}